// Net_57690000720016
// MI455X (gfx1250) — compile-verified
//
#include <hip/hip_runtime.h>
#include <math.h>

typedef __attribute__((ext_vector_type(2))) float v2f;
typedef __attribute__((ext_vector_type(8))) float v8f;
typedef __attribute__((ext_vector_type(4))) unsigned int v4u;
typedef __attribute__((ext_vector_type(8))) int v8i;
typedef __attribute__((ext_vector_type(4))) int v4i;

#define NEG_SLOPE 0.2f

// ---------------------------------------------------------------- utilities

__device__ __forceinline__ void atomicMaxF(float* addr, float val) {
    unsigned int* ai = (unsigned int*)addr;
    unsigned int old = __float_as_uint(*addr);
    while (__uint_as_float(old) < val) {
        unsigned int assumed = old;
        old = atomicCAS(ai, assumed, __float_as_uint(val));
        if (old == assumed) break;
    }
}

__device__ __forceinline__ void edge_nodes(const long long* __restrict__ ei,
                                           long long n_edges, long long e,
                                           int& s, int& d) {
    if (e < n_edges) {
        s = (int)ei[e];
        d = (int)ei[n_edges + e];
    } else {
        s = d = (int)(e - n_edges);   // self loop
    }
}

__global__ void fillk(float* __restrict__ p, float v, long long n) {
    long long i = (long long)blockIdx.x * blockDim.x + threadIdx.x;
    long long stride = (long long)gridDim.x * blockDim.x;
    for (; i < n; i += stride) p[i] = v;
}

// ---------------------------------------------------------------- GEMM1 + attention logits
// h[N,128] = x[N,256] @ W1[256,128]; per wave: one 16x16 tile == one head.
// x-tile staged in LDS by the Tensor Data Mover (async DMA) with row padding
// 256 -> 260 dwords (TDM pad: interval 256 dwords, amount 4 dwords) so the
// WMMA A-fragment ds_loads are bank-conflict free. OOB rows read as zero.
// Also fuses s_src[N,8], s_dst[N,8] via 16-lane shuffle reductions.

#define XS 260  // padded LDS row stride (floats)

__global__ __launch_bounds__(256) void gemm1_wmma(
    const float* __restrict__ x, const float* __restrict__ W1,
    const float* __restrict__ a1s, const float* __restrict__ a1d,
    float* __restrict__ h, float* __restrict__ ssrc, float* __restrict__ sdst,
    int n_nodes)
{
    __shared__ float xs[16 * XS];
    const int tid  = threadIdx.x;
    const int row0 = blockIdx.x * 16;

    // ---- TDM async tensor load: 16x256 f32 tile, global -> LDS
    if (tid < 32) {
        unsigned long long gaddr =
            (unsigned long long)(uintptr_t)x + (unsigned long long)row0 * 1024ull;
        unsigned int lds_off   = (unsigned int)(uintptr_t)xs;   // LDS byte offset
        unsigned int rows_left = (unsigned int)(n_nodes - row0); // tail rows -> OOB zeros

        v4u g0;
        g0[0] = 1u;                                  // count=1 valid descriptor
        g0[1] = lds_off;                             // lds_addr
        g0[2] = (unsigned int)gaddr;                 // global_addr[31:0]
        g0[3] = (unsigned int)((gaddr >> 32) & 0x1FFFFFFull) | (2u << 30); // addr[56:32] | type=2

        v8i g1;
        g1[0] = (int)((2u << 16)        // data_size = 4 bytes
                    | (1u << 20)        // pad_enable
                    | (7u << 22)        // pad_interval: every 256 dwords
                    | (3u << 25));      // pad_amount: 4 dwords  (stride 260)
        g1[1] = (int)(256u << 16);                               // tensor_dim0 = 256
        g1[2] = (int)((rows_left & 0xFFFFu) << 16);              // tensor_dim1[15:0]
        g1[3] = (int)(((rows_left >> 16) & 0xFFFFu)              // tensor_dim1[31:16]
                    | (256u << 16));                             // tile_dim0 = 256
        g1[4] = 16;                                              // tile_dim1 = 16 (tile_dim2=0)
        g1[5] = 256;                                             // tensor_dim0_stride = 256
        g1[6] = 0;
        g1[7] = 0;

        v4i gz  = {0, 0, 0, 0};                                  // 2D: groups 2/3 unused
        v8i gz8 = {0, 0, 0, 0, 0, 0, 0, 0};
        __builtin_amdgcn_tensor_load_to_lds(g0, g1, gz, gz, gz8, 0);
        __builtin_amdgcn_s_wait_tensorcnt(0);
    }
    __syncthreads();

    const int wave = tid >> 5;         // head 0..7, cols [16w, 16w+16)
    const int lane = tid & 31;
    const int half = lane >> 4;        // lane half selects K pair {0,1} vs {2,3}
    const int l16  = lane & 15;
    const int col0 = wave * 16;

    v8f acc = {};
    #pragma unroll 4
    for (int k = 0; k < 256; k += 4) {
        const int ka = k + 2 * half;
        v2f a, b;
        a.x = xs[l16 * XS + ka];
        a.y = xs[l16 * XS + ka + 1];
        b.x = W1[(size_t)ka * 128 + col0 + l16];
        b.y = W1[(size_t)(ka + 1) * 128 + col0 + l16];
        acc = __builtin_amdgcn_wmma_f32_16x16x4_f32(false, a, false, b,
                                                    (short)0, acc, false, false);
    }

    const float was = a1s[wave * 16 + l16];
    const float wad = a1d[wave * 16 + l16];
    #pragma unroll
    for (int i = 0; i < 8; i++) {
        const int r = row0 + i + 8 * half;         // C/D layout: row = i + 8*half
        const float v = acc[i];
        if (r < n_nodes) h[(size_t)r * 128 + col0 + l16] = v;
        float vs = v * was, vd = v * wad;
        #pragma unroll
        for (int off = 8; off; off >>= 1) {        // reduce within each 16-lane half
            vs += __shfl_xor(vs, off, 32);
            vd += __shfl_xor(vd, off, 32);
        }
        if (l16 == 0 && r < n_nodes) {
            ssrc[r * 8 + wave] = vs;
            sdst[r * 8 + wave] = vd;
        }
    }
}

// ---------------------------------------------------------------- layer-1 edge passes (H=8)

__global__ void edge1_logit_max(const long long* __restrict__ ei, long long n_edges,
                                long long n_tot, const float* __restrict__ ssrc,
                                const float* __restrict__ sdst,
                                float* __restrict__ e1, float* __restrict__ m1)
{
    long long idx = (long long)blockIdx.x * blockDim.x + threadIdx.x;
    if (idx >= n_tot * 8) return;
    long long e = idx >> 3;
    int hh = (int)(idx & 7);
    int s, d; edge_nodes(ei, n_edges, e, s, d);
    float v = ssrc[(size_t)s * 8 + hh] + sdst[(size_t)d * 8 + hh];
    v = (v > 0.f) ? v : NEG_SLOPE * v;
    e1[idx] = v;
    atomicMaxF(&m1[(size_t)d * 8 + hh], v);
}

__global__ void edge1_exp_sum(const long long* __restrict__ ei, long long n_edges,
                              long long n_tot, const float* __restrict__ m1,
                              float* __restrict__ e1, float* __restrict__ d1)
{
    long long idx = (long long)blockIdx.x * blockDim.x + threadIdx.x;
    if (idx >= n_tot * 8) return;
    long long e = idx >> 3;
    int hh = (int)(idx & 7);
    int s, d; edge_nodes(ei, n_edges, e, s, d);
    float ex = __expf(e1[idx] - m1[(size_t)d * 8 + hh]);
    e1[idx] = ex;
    atomicAdd(&d1[(size_t)d * 8 + hh], ex);
}

__global__ void edge1_aggregate(const long long* __restrict__ ei, long long n_edges,
                                long long n_tot, const float* __restrict__ e1,
                                const float* __restrict__ d1,
                                const float* __restrict__ h, float* __restrict__ agg)
{
    long long idx = (long long)blockIdx.x * blockDim.x + threadIdx.x;
    if (idx >= n_tot * 8) return;
    long long e = idx >> 3;
    int hh = (int)(idx & 7);
    int s, d; edge_nodes(ei, n_edges, e, s, d);
    float alpha = e1[idx] / (d1[(size_t)d * 8 + hh] + 1e-16f);
    const float4* hp = (const float4*)(h + (size_t)s * 128 + hh * 16);
    float* ap = agg + (size_t)d * 128 + hh * 16;
    #pragma unroll
    for (int q = 0; q < 4; q++) {
        float4 hv = hp[q];
        atomicAdd(ap + 4 * q + 0, alpha * hv.x);
        atomicAdd(ap + 4 * q + 1, alpha * hv.y);
        atomicAdd(ap + 4 * q + 2, alpha * hv.z);
        atomicAdd(ap + 4 * q + 3, alpha * hv.w);
    }
}

__global__ void bias_elu(float* __restrict__ agg, const float* __restrict__ b1,
                         long long n)
{
    long long i = (long long)blockIdx.x * blockDim.x + threadIdx.x;
    if (i >= n) return;
    float v = agg[i] + b1[i & 127];
    agg[i] = (v > 0.f) ? v : expm1f(v);
}

// ---------------------------------------------------------------- GEMM2 (47 classes, pad to 48)

#define HS 132  // padded LDS row stride for the 16x128 tile

__global__ __launch_bounds__(96) void gemm2_wmma(
    const float* __restrict__ hin, const float* __restrict__ W2,
    float* __restrict__ t, int n_nodes)
{
    __shared__ float hs[16 * HS];
    const int tid  = threadIdx.x;
    const int row0 = blockIdx.x * 16;

    for (int i = tid; i < 16 * 128; i += 96) {
        int r = i >> 7, c = i & 127;
        int gr = row0 + r;
        hs[r * HS + c] = (gr < n_nodes) ? hin[(size_t)gr * 128 + c] : 0.f;
    }
    __syncthreads();

    const int wave = tid >> 5;       // col tile 0..2
    const int lane = tid & 31;
    const int half = lane >> 4;
    const int l16  = lane & 15;
    const int col  = wave * 16 + l16;

    v8f acc = {};
    #pragma unroll 4
    for (int k = 0; k < 128; k += 4) {
        const int ka = k + 2 * half;
        v2f a, b;
        a.x = hs[l16 * HS + ka];
        a.y = hs[l16 * HS + ka + 1];
        b.x = (col < 47) ? W2[(size_t)ka * 47 + col] : 0.f;
        b.y = (col < 47) ? W2[(size_t)(ka + 1) * 47 + col] : 0.f;
        acc = __builtin_amdgcn_wmma_f32_16x16x4_f32(false, a, false, b,
                                                    (short)0, acc, false, false);
    }

    if (col < 47) {
        #pragma unroll
        for (int i = 0; i < 8; i++) {
            const int r = row0 + i + 8 * half;
            if (r < n_nodes) t[(size_t)r * 47 + col] = acc[i];
        }
    }
}

__global__ void s2_kernel(const float* __restrict__ t, const float* __restrict__ a2s,
                          const float* __restrict__ a2d, float* __restrict__ s2s,
                          float* __restrict__ s2d, int n_nodes)
{
    int n = blockIdx.x * blockDim.x + threadIdx.x;
    if (n >= n_nodes) return;
    const float* r = t + (size_t)n * 47;
    float ss = 0.f, sd = 0.f;
    for (int c = 0; c < 47; c++) { ss += r[c] * a2s[c]; sd += r[c] * a2d[c]; }
    s2s[n] = ss; s2d[n] = sd;
}

// ---------------------------------------------------------------- layer-2 edge passes (H=1)

__global__ void edge2_logit_max(const long long* __restrict__ ei, long long n_edges,
                                long long n_tot, const float* __restrict__ s2s,
                                const float* __restrict__ s2d,
                                float* __restrict__ e2, float* __restrict__ m2)
{
    long long e = (long long)blockIdx.x * blockDim.x + threadIdx.x;
    if (e >= n_tot) return;
    int s, d; edge_nodes(ei, n_edges, e, s, d);
    float v = s2s[s] + s2d[d];
    v = (v > 0.f) ? v : NEG_SLOPE * v;
    e2[e] = v;
    atomicMaxF(&m2[d], v);
}

__global__ void edge2_exp_sum(const long long* __restrict__ ei, long long n_edges,
                              long long n_tot, const float* __restrict__ m2,
                              float* __restrict__ e2, float* __restrict__ d2)
{
    long long e = (long long)blockIdx.x * blockDim.x + threadIdx.x;
    if (e >= n_tot) return;
    int s, d; edge_nodes(ei, n_edges, e, s, d);
    float ex = __expf(e2[e] - m2[d]);
    e2[e] = ex;
    atomicAdd(&d2[d], ex);
}

__global__ void edge2_alpha(const long long* __restrict__ ei, long long n_edges,
                            long long n_tot, const float* __restrict__ d2,
                            float* __restrict__ e2)
{
    long long e = (long long)blockIdx.x * blockDim.x + threadIdx.x;
    if (e >= n_tot) return;
    int s, d; edge_nodes(ei, n_edges, e, s, d);
    e2[e] = e2[e] / (d2[d] + 1e-16f);
}

__global__ void edge2_aggregate(const long long* __restrict__ ei, long long n_edges,
                                long long n_tot, const float* __restrict__ alpha2,
                                const float* __restrict__ t, float* __restrict__ agg2)
{
    long long idx = (long long)blockIdx.x * blockDim.x + threadIdx.x;
    if (idx >= n_tot * 64) return;
    long long e = idx >> 6;
    int c = (int)(idx & 63);
    if (c >= 47) return;
    int s, d; edge_nodes(ei, n_edges, e, s, d);
    atomicAdd(&agg2[(size_t)d * 47 + c], alpha2[e] * t[(size_t)s * 47 + c]);
}

// ---------------------------------------------------------------- log-softmax (wave per node)

__global__ __launch_bounds__(256) void logsoftmax_k(const float* __restrict__ agg2,
                                                    const float* __restrict__ b2,
                                                    float* __restrict__ out, int n_nodes)
{
    const int wave = threadIdx.x >> 5;
    const int lane = threadIdx.x & 31;
    const int n = blockIdx.x * 8 + wave;
    if (n >= n_nodes) return;
    const float* r = agg2 + (size_t)n * 47;
    float v0 = (lane < 47)      ? r[lane]      + b2[lane]      : -__builtin_inff();
    float v1 = (lane + 32 < 47) ? r[lane + 32] + b2[lane + 32] : -__builtin_inff();
    float m = fmaxf(v0, v1);
    #pragma unroll
    for (int off = 16; off; off >>= 1) m = fmaxf(m, __shfl_xor(m, off, 32));
    float s = 0.f;
    if (lane < 47)      s += __expf(v0 - m);
    if (lane + 32 < 47) s += __expf(v1 - m);
    #pragma unroll
    for (int off = 16; off; off >>= 1) s += __shfl_xor(s, off, 32);
    float lse = m + __logf(s);
    if (lane < 47)      out[(size_t)n * 47 + lane]      = v0 - lse;
    if (lane + 32 < 47) out[(size_t)n * 47 + lane + 32] = v1 - lse;
}

// ---------------------------------------------------------------- launch

extern "C" void kernel_launch(void* const* d_in, const int* in_sizes, int n_in,
                              void* d_out, int out_size, void* d_ws, size_t ws_size,
                              hipStream_t stream) {
    const float*     x   = (const float*)d_in[0];
    const long long* ei  = (const long long*)d_in[1];
    const float*     W1  = (const float*)d_in[2];
    const float*     a1s = (const float*)d_in[3];
    const float*     a1d = (const float*)d_in[4];
    const float*     b1  = (const float*)d_in[5];
    const float*     W2  = (const float*)d_in[6];
    const float*     a2s = (const float*)d_in[7];
    const float*     a2d = (const float*)d_in[8];
    const float*     b2  = (const float*)d_in[9];
    float* out = (float*)d_out;

    const long long n_nodes = in_sizes[0] / 256;
    const long long n_edges = in_sizes[1] / 2;
    const long long n_tot   = n_edges + n_nodes;   // + self loops
    (void)n_in; (void)out_size; (void)ws_size;

    // workspace carve-up (floats, 64-element aligned)
    float* ws = (float*)d_ws;
    size_t off = 0;
    auto take = [&](size_t n) { float* p = ws + off; off += (n + 63) & ~(size_t)63; return p; };
    float* h1   = take((size_t)n_nodes * 128);
    float* s1s  = take((size_t)n_nodes * 8);
    float* s1d  = take((size_t)n_nodes * 8);
    float* m1   = take((size_t)n_nodes * 8);
    float* d1   = take((size_t)n_nodes * 8);
    float* e1   = take((size_t)n_tot * 8);
    float* agg1 = take((size_t)n_nodes * 128);
    float* t2   = take((size_t)n_nodes * 47);
    float* s2s  = take((size_t)n_nodes);
    float* s2d  = take((size_t)n_nodes);
    float* m2   = take((size_t)n_nodes);
    float* d2   = take((size_t)n_nodes);
    float* e2   = take((size_t)n_tot);
    float* agg2 = take((size_t)n_nodes * 47);

    const float NEG_INF = -__builtin_inff();
    const int FB = 2048;

    // ---- init accumulators (every call; harness does not re-poison)
    fillk<<<FB, 256, 0, stream>>>(m1,   NEG_INF, n_nodes * 8);
    fillk<<<FB, 256, 0, stream>>>(d1,   0.f,     n_nodes * 8);
    fillk<<<FB, 256, 0, stream>>>(agg1, 0.f,     n_nodes * 128);
    fillk<<<FB, 256, 0, stream>>>(m2,   NEG_INF, n_nodes);
    fillk<<<FB, 256, 0, stream>>>(d2,   0.f,     n_nodes);
    fillk<<<FB, 256, 0, stream>>>(agg2, 0.f,     n_nodes * 47);

    // ---- layer 1
    int g1 = (int)((n_nodes + 15) / 16);
    gemm1_wmma<<<g1, 256, 0, stream>>>(x, W1, a1s, a1d, h1, s1s, s1d, (int)n_nodes);

    int ge8 = (int)((n_tot * 8 + 255) / 256);
    edge1_logit_max<<<ge8, 256, 0, stream>>>(ei, n_edges, n_tot, s1s, s1d, e1, m1);
    edge1_exp_sum  <<<ge8, 256, 0, stream>>>(ei, n_edges, n_tot, m1, e1, d1);
    edge1_aggregate<<<ge8, 256, 0, stream>>>(ei, n_edges, n_tot, e1, d1, h1, agg1);

    int gb = (int)((n_nodes * 128 + 255) / 256);
    bias_elu<<<gb, 256, 0, stream>>>(agg1, b1, n_nodes * 128);

    // ---- layer 2
    gemm2_wmma<<<g1, 96, 0, stream>>>(agg1, W2, t2, (int)n_nodes);
    int gn = (int)((n_nodes + 255) / 256);
    s2_kernel<<<gn, 256, 0, stream>>>(t2, a2s, a2d, s2s, s2d, (int)n_nodes);

    int ge1 = (int)((n_tot + 255) / 256);
    edge2_logit_max<<<ge1, 256, 0, stream>>>(ei, n_edges, n_tot, s2s, s2d, e2, m2);
    edge2_exp_sum  <<<ge1, 256, 0, stream>>>(ei, n_edges, n_tot, m2, e2, d2);
    edge2_alpha    <<<ge1, 256, 0, stream>>>(ei, n_edges, n_tot, d2, e2);
    int ge64 = (int)((n_tot * 64 + 255) / 256);
    edge2_aggregate<<<ge64, 256, 0, stream>>>(ei, n_edges, n_tot, e2, t2, agg2);

    // ---- output
    int gl = (int)((n_nodes + 7) / 8);
    logsoftmax_k<<<gl, 256, 0, stream>>>(agg2, b2, out, (int)n_nodes);
}